// SS_76355928588920
// MI455X (gfx1250) — compile-verified
//
#include <hip/hip_runtime.h>
#include <stdint.h>

// ---------------------------------------------------------------------------
// SSIM (mean) for (32,3,512,512) fp32 pairs on gfx1250.
// Separable 11-tap Gaussian conv expressed as banded matmuls on
// V_WMMA_F32_16X16X4_F32 (full fp32 precision on the matrix pipe).
// Input staging uses GLOBAL_LOAD_ASYNC_TO_LDS_B32 (ASYNCcnt path) for
// interior tiles; border tiles use a clamped VGPR path (wave-uniform branch).
// ---------------------------------------------------------------------------

typedef __attribute__((ext_vector_type(2))) float v2f;
typedef __attribute__((ext_vector_type(8))) float v8f;

#define H_IMG 512
#define W_IMG 512
#define N_PLANES 96            // 32 * 3
#define TILES_PER_PLANE 1024   // 32 x 32 tiles of 16x16
#define N_TILES (N_PLANES * TILES_PER_PLANE)   // 98304
#define WAVES_PER_BLOCK 4
#define N_BLOCKS (N_TILES / WAVES_PER_BLOCK)   // 24576
#define SSIM_C2 0.0009f        // (0.03*1.0)^2
#define INV_COUNT (1.0f / 25165824.0f)         // 1/(32*3*512*512)

// Normalized gaussian window, sigma=1.5, 11 taps.
__device__ __constant__ float GW[11] = {
    0.0010284f, 0.0075988f, 0.0360008f, 0.1093608f, 0.2130056f,
    0.2660117f,
    0.2130056f, 0.1093608f, 0.0360008f, 0.0075988f, 0.0010284f};

__device__ __forceinline__ float band(int d) {
    int dc = d < 0 ? 0 : (d > 10 ? 10 : d);
    float w = GW[dc];
    return ((unsigned)d <= 10u) ? w : 0.0f;
}

__device__ __forceinline__ v8f wmma_f32_4(v2f a, v2f b, v8f c) {
    // D = A(16x4 f32) * B(4x16 f32) + C(16x16 f32)
    return __builtin_amdgcn_wmma_f32_16x16x4_f32(
        false, a, false, b, (short)0, c, false, false);
}

// Async global -> LDS copy, 4 bytes per active lane. Tracked by ASYNCcnt.
__device__ __forceinline__ void async_copy_b32(uint32_t lds_addr, const float* gaddr) {
    asm volatile("global_load_async_to_lds_b32 %0, %1, off"
                 :: "v"(lds_addr), "v"((uint64_t)(uintptr_t)gaddr)
                 : "memory");
}

// Per-wave LDS layout (floats):
//   [0      .. 727 ]  img1 tile  26 rows x 28 cols
//   [728    .. 1455]  img2 tile  26 rows x 28 cols
//   [1456   .. 3695]  H[5][16 cols][28 rows]  (rows 26,27 zero) column-major
#define LDS_PER_WAVE 3696

__global__ __launch_bounds__(WAVES_PER_BLOCK * 32)
void ssim_tile_kernel(const float* __restrict__ img1,
                      const float* __restrict__ img2,
                      float* __restrict__ partials) {
    __shared__ float lds[WAVES_PER_BLOCK][LDS_PER_WAVE];
    __shared__ float blkSum[WAVES_PER_BLOCK];

    const int tid    = threadIdx.x;
    const int wave   = tid >> 5;
    const int lane   = tid & 31;
    const int lane16 = lane & 15;
    const int half   = lane >> 4;          // 0: lanes 0-15, 1: lanes 16-31

    float* Win = lds[wave];                // inputs
    float* Hq  = lds[wave] + 1456;         // H, column-major per map

    const int tileId = blockIdx.x * WAVES_PER_BLOCK + wave;
    const int plane  = tileId >> 10;
    const int tin    = tileId & 1023;
    const int trow   = tin >> 5;
    const int tcol   = tin & 31;
    const int r0     = trow * 16 - 5;      // top-left of 26x26 halo region
    const int c0     = tcol * 16 - 5;

    const float* p1 = img1 + (size_t)plane * (H_IMG * W_IMG);
    const float* p2 = img2 + (size_t)plane * (H_IMG * W_IMG);

    // ---- stage 26x28 input tiles into LDS ----
    // Interior tiles: straight async copy (pad cols 26,27 carry real data but
    // their band weights are exactly zero, so they contribute nothing).
    // Border tiles: clamped VGPR path with explicit zero padding.
    const bool interior = (r0 >= 0) & (c0 >= 0) &
                          (r0 + 26 <= H_IMG) & (c0 + 28 <= W_IMG);
    if (interior) {                         // wave-uniform branch
        const uint32_t ldsbase = (uint32_t)(uintptr_t)Win;
        for (int idx = lane; idx < 728; idx += 32) {
            int rr = idx / 28, cc = idx - rr * 28;
            int off = (r0 + rr) * W_IMG + (c0 + cc);
            async_copy_b32(ldsbase + (uint32_t)idx * 4u, p1 + off);
            async_copy_b32(ldsbase + 2912u + (uint32_t)idx * 4u, p2 + off);
        }
        asm volatile("s_wait_asynccnt 0x0" ::: "memory");
    } else {
        for (int idx = lane; idx < 728; idx += 32) {
            int rr = idx / 28, cc = idx - rr * 28;
            int gr = r0 + rr, gc = c0 + cc;
            bool ok = (cc < 26) & (gr >= 0) & (gr < H_IMG) & (gc >= 0) & (gc < W_IMG);
            int grc = gr < 0 ? 0 : (gr > H_IMG - 1 ? H_IMG - 1 : gr);
            int gcc = gc < 0 ? 0 : (gc > W_IMG - 1 ? W_IMG - 1 : gc);
            int off = grc * W_IMG + gcc;
            float a = p1[off], b = p2[off];
            Win[idx]       = ok ? a : 0.0f;
            Win[idx + 728] = ok ? b : 0.0f;
        }
    }

    // ---- banded weight chunks: identical per-lane value serves as
    //      B (pass 1: Wh[k][n]=w[k-n]) and A (pass 2: Wv[m][k]=w[k-m]) ----
    v2f Wc[7];
#pragma unroll
    for (int i = 0; i < 7; ++i) {
        int kA = 4 * i + 2 * half;
        Wc[i].x = band(kA - lane16);
        Wc[i].y = band(kA + 1 - lane16);
    }

    const v8f vzero = {0.f, 0.f, 0.f, 0.f, 0.f, 0.f, 0.f, 0.f};

    // ---- pass 1: horizontal conv  H_q(26x16) = In_q(26x28) x Wh(28x16) ----
    v8f Hacc[5][2];
#pragma unroll
    for (int q = 0; q < 5; ++q) {
        Hacc[q][0] = vzero;
        Hacc[q][1] = vzero;
    }

#pragma unroll
    for (int g = 0; g < 2; ++g) {
#pragma unroll
        for (int i = 0; i < 7; ++i) {
            int rA = g * 16 + lane16;          // A row (M)
            int kA = 4 * i + 2 * half;         // A col pair (K)
            int o  = rA * 28 + kA;
            v2f a, b;
            a.x = Win[o];       a.y = Win[o + 1];
            b.x = Win[o + 728]; b.y = Win[o + 729];
            v2f aa = a * a, bb = b * b, ab = a * b;
            Hacc[0][g] = wmma_f32_4(a,  Wc[i], Hacc[0][g]);
            Hacc[1][g] = wmma_f32_4(b,  Wc[i], Hacc[1][g]);
            Hacc[2][g] = wmma_f32_4(aa, Wc[i], Hacc[2][g]);
            Hacc[3][g] = wmma_f32_4(bb, Wc[i], Hacc[3][g]);
            Hacc[4][g] = wmma_f32_4(ab, Wc[i], Hacc[4][g]);
        }
    }

    // ---- spill H to LDS column-major: Hq[q*448 + col*28 + row] ----
#pragma unroll
    for (int q = 0; q < 5; ++q) {
#pragma unroll
        for (int g = 0; g < 2; ++g) {
#pragma unroll
            for (int v = 0; v < 8; ++v) {
                int row = g * 16 + v + 8 * half;   // D layout: M = v (+8 hi half)
                if (row < 26)
                    Hq[q * 448 + lane16 * 28 + row] = Hacc[q][g][v];
            }
        }
        Hq[q * 448 + lane16 * 28 + 26 + half] = 0.0f;   // zero K-pad rows
    }

    // ---- pass 2: vertical conv  Out_q(16x16) = Wv(16x28) x H_q(28x16) ----
    v8f O[5];
#pragma unroll
    for (int q = 0; q < 5; ++q) O[q] = vzero;

#pragma unroll
    for (int q = 0; q < 5; ++q) {
#pragma unroll
        for (int i = 0; i < 7; ++i) {
            int kA = 4 * i + 2 * half;
            int o  = q * 448 + lane16 * 28 + kA;   // B: lane holds H[kA..kA+1][n]
            v2f hb;
            hb.x = Hq[o];
            hb.y = Hq[o + 1];
            O[q] = wmma_f32_4(Wc[i], hb, O[q]);
        }
    }

    // ---- pointwise SSIM map + reduction (8 pixels per lane) ----
    float sum = 0.0f;
#pragma unroll
    for (int v = 0; v < 8; ++v) {
        float mu1 = O[0][v], mu2 = O[1][v];
        float s11 = O[2][v] - mu1 * mu1;
        float s22 = O[3][v] - mu2 * mu2;
        float s12 = O[4][v] - mu1 * mu2;
        float s1 = __builtin_sqrtf(__builtin_fabsf(s11));
        float s2 = __builtin_sqrtf(__builtin_fabsf(s22));
        sum += (s12 + SSIM_C2) / (s1 * s2 + SSIM_C2);
    }
#pragma unroll
    for (int off = 16; off > 0; off >>= 1)
        sum += __shfl_xor(sum, off, 32);

    if (lane == 0) blkSum[wave] = sum;
    __syncthreads();
    if (tid == 0) {
        float t = 0.0f;
#pragma unroll
        for (int w = 0; w < WAVES_PER_BLOCK; ++w) t += blkSum[w];
        partials[blockIdx.x] = t;
    }
}

__global__ __launch_bounds__(256)
void ssim_reduce_kernel(const float* __restrict__ partials,
                        float* __restrict__ out, int n) {
    __shared__ float s[256];
    float t = 0.0f;
    for (int i = threadIdx.x; i < n; i += 256) t += partials[i];
    s[threadIdx.x] = t;
    __syncthreads();
    for (int st = 128; st > 0; st >>= 1) {
        if ((int)threadIdx.x < st) s[threadIdx.x] += s[threadIdx.x + st];
        __syncthreads();
    }
    if (threadIdx.x == 0) out[0] = s[0] * INV_COUNT;
}

extern "C" void kernel_launch(void* const* d_in, const int* in_sizes, int n_in,
                              void* d_out, int out_size, void* d_ws, size_t ws_size,
                              hipStream_t stream) {
    const float* img1 = (const float*)d_in[0];
    const float* img2 = (const float*)d_in[1];
    float* out      = (float*)d_out;
    float* partials = (float*)d_ws;   // N_BLOCKS floats = 96 KB

    ssim_tile_kernel<<<N_BLOCKS, WAVES_PER_BLOCK * 32, 0, stream>>>(img1, img2, partials);
    ssim_reduce_kernel<<<1, 256, 0, stream>>>(partials, out, N_BLOCKS);
}